// ArrayMultiplier4x4_Strict_23407571764172
// MI455X (gfx1250) — compile-verified
//
#include <hip/hip_runtime.h>
#include <cstdint>

// ---------------------------------------------------------------------------
// 4-bit x 4-bit spiking array multiplier, compiled down to its exact Boolean
// semantics: out[r, 0..7] = bits of (a4(r) * b4(r)), LSB first, as 0.0/1.0 f32.
//
// Bandwidth-bound: 64 B/row * 4Mi rows = 256 MiB -> ~11.5 us at 23.3 TB/s.
// Strategy: b128 loads/stores, async global->LDS double-buffered prefetch
// pipeline synchronized with ASYNCcnt (CDNA5), non-temporal output stores.
// ---------------------------------------------------------------------------

#define GLOBAL_AS __attribute__((address_space(1)))
#define LDS_AS    __attribute__((address_space(3)))

typedef float v4f __attribute__((ext_vector_type(4)));
typedef int   v4i __attribute__((vector_size(16)));   // matches builtin param type

constexpr int BLOCK = 256;           // 8 wave32 waves per workgroup
constexpr int MAX_BLOCKS = 2048;     // persistent blocks; >=2 tiles/block for pipelining

#if defined(__HIP_DEVICE_COMPILE__) && \
    __has_builtin(__builtin_amdgcn_global_load_async_to_lds_b128) && \
    __has_builtin(__builtin_amdgcn_s_wait_asynccnt)
#define USE_ASYNC_LDS 1
#else
#define USE_ASYNC_LDS 0
#endif

__global__ __launch_bounds__(BLOCK) void snn_mul4x4_kernel(
    const float* __restrict__ A, const float* __restrict__ B,
    float* __restrict__ O, long long rows, int numTiles)
{
    const v4f* __restrict__ A4 = (const v4f*)A;   // one row = 16 B
    const v4f* __restrict__ B4 = (const v4f*)B;   // one row = 16 B
    v4f* __restrict__ O4 = (v4f*)O;               // one row = 32 B (2 x v4f)

    const int tid    = (int)threadIdx.x;
    const int stride = (int)gridDim.x;

#if USE_ASYNC_LDS
    // Double-buffered staging tiles. Each wave only ever reads back the LDS
    // slots it issued async loads for itself, so ASYNCcnt (per-wave, in-order
    // for async loads) is the only synchronization needed — no barriers.
    __shared__ v4f lA[2][BLOCK];
    __shared__ v4f lB[2][BLOCK];

    auto issue = [&](int k) {
        const int tile = (int)blockIdx.x + k * stride;
        if (tile < numTiles) {
            const long long r = (long long)tile * BLOCK + tid;
            if (r < rows) {
                const int s = k & 1;
                __builtin_amdgcn_global_load_async_to_lds_b128(
                    (GLOBAL_AS v4i*)(uintptr_t)(A4 + r),
                    (LDS_AS v4i*)(uint32_t)(uintptr_t)&lA[s][tid],
                    /*offset=*/0, /*cpol=*/0);
                __builtin_amdgcn_global_load_async_to_lds_b128(
                    (GLOBAL_AS v4i*)(uintptr_t)(B4 + r),
                    (LDS_AS v4i*)(uint32_t)(uintptr_t)&lB[s][tid],
                    /*offset=*/0, /*cpol=*/0);
            }
        }
    };

    issue(0);  // prologue: stage 0 in flight
#endif

    for (int k = 0;; ++k) {
        const int tile = (int)blockIdx.x + k * stride;
        if (tile >= numTiles) break;

#if USE_ASYNC_LDS
        // Prefetch next tile, then wait until this tile's 2 async loads (the
        // oldest outstanding) have landed in LDS. Async loads complete in
        // order, so ASYNCcnt <= 2 leaves exactly the next tile in flight.
        issue(k + 1);
        if ((int)blockIdx.x + (k + 1) * stride < numTiles) {
            __builtin_amdgcn_s_wait_asynccnt(2);
        } else {
            __builtin_amdgcn_s_wait_asynccnt(0);
        }
#endif

        const long long r = (long long)tile * BLOCK + tid;
        if (r < rows) {
            v4f a, b;
#if USE_ASYNC_LDS
            a = lA[k & 1][tid];      // ds_load_b128
            b = lB[k & 1][tid];      // ds_load_b128
#else
            a = A4[r];               // global_load_b128
            b = B4[r];
            // CDNA5 prefetch of the next tile this block will touch.
            const int ntile = (int)blockIdx.x + (k + 1) * stride;
            if (ntile < numTiles) {
                const long long nr = (long long)ntile * BLOCK + tid;
                __builtin_prefetch((const void*)(A4 + nr), 0, 0);
                __builtin_prefetch((const void*)(B4 + nr), 0, 0);
            }
#endif
            // Inputs are exactly 0.0f or 1.0f -> exact integer recovery.
            const int ia = (int)a[0] | ((int)a[1] << 1) | ((int)a[2] << 2) | ((int)a[3] << 3);
            const int ib = (int)b[0] | ((int)b[1] << 1) | ((int)b[2] << 2) | ((int)b[3] << 3);
            const int p  = ia * ib;   // 0..225, fits in 8 bits

            v4f lo, hi;
            lo[0] = (float)( p       & 1);
            lo[1] = (float)((p >> 1) & 1);
            lo[2] = (float)((p >> 2) & 1);
            lo[3] = (float)((p >> 3) & 1);
            hi[0] = (float)((p >> 4) & 1);
            hi[1] = (float)((p >> 5) & 1);
            hi[2] = (float)((p >> 6) & 1);
            hi[3] = (float)((p >> 7) & 1);

            // Output is written once and never re-read on device: NT stores.
            __builtin_nontemporal_store(lo, O4 + 2 * r);
            __builtin_nontemporal_store(hi, O4 + 2 * r + 1);
        }
    }
}

extern "C" void kernel_launch(void* const* d_in, const int* in_sizes, int n_in,
                              void* d_out, int out_size, void* d_ws, size_t ws_size,
                              hipStream_t stream) {
    (void)n_in; (void)out_size; (void)d_ws; (void)ws_size;

    const float* A = (const float*)d_in[0];
    const float* B = (const float*)d_in[1];
    float* O = (float*)d_out;

    const long long rows = (long long)in_sizes[0] / 4;   // A is [N, 4]
    int numTiles = (int)((rows + BLOCK - 1) / BLOCK);
    if (numTiles < 1) numTiles = 0;

    int blocks = numTiles;
    if (blocks > MAX_BLOCKS) blocks = MAX_BLOCKS;   // persistent blocks, >=2 tiles each
    if (blocks < 1) blocks = 1;

    snn_mul4x4_kernel<<<dim3(blocks), dim3(BLOCK), 0, stream>>>(A, B, O, rows, numTiles);
}